// gcn_lm_14250701488890
// MI455X (gfx1250) — compile-verified
//
#include <hip/hip_runtime.h>
#include <hip/hip_bf16.h>

typedef float v2f __attribute__((ext_vector_type(2)));
typedef float v8f __attribute__((ext_vector_type(8)));

#define LN_EPS 1e-5f

// ---------------- LayerNorm: one wave (32 lanes) per 256-wide row ----------------
__global__ void gcn_ln_kernel(const float* __restrict__ x, const float* __restrict__ gamma,
                              const float* __restrict__ beta, float* __restrict__ out, int n) {
  int row = blockIdx.x * (blockDim.x >> 5) + (threadIdx.x >> 5);
  if (row >= n) return;
  int lane = threadIdx.x & 31;
  const float* xr = x + (size_t)row * 256;
  float v[8];
  float s = 0.f;
#pragma unroll
  for (int i = 0; i < 8; ++i) { v[i] = xr[i * 32 + lane]; s += v[i]; }
#pragma unroll
  for (int o = 16; o > 0; o >>= 1) s += __shfl_xor(s, o, 32);
  float mu = s * (1.f / 256.f);
  float var = 0.f;
#pragma unroll
  for (int i = 0; i < 8; ++i) { float d = v[i] - mu; var += d * d; }
#pragma unroll
  for (int o = 16; o > 0; o >>= 1) var += __shfl_xor(var, o, 32);
  float rs = rsqrtf(var * (1.f / 256.f) + LN_EPS);
  float* orow = out + (size_t)row * 256;
#pragma unroll
  for (int i = 0; i < 8; ++i) {
    int f = i * 32 + lane;
    orow[f] = (v[i] - mu) * rs * gamma[f] + beta[f];
  }
}

// ---------------- Dense GEMM via V_WMMA_F32_16X16X4_F32 ----------------
// Block = 128 threads (4 waves). Block computes rows [16*bx, 16*bx+16) x cols [64*by, 64*by+64).
// A tile (16 x K) staged in LDS (padded stride K+4 -> conflict-free fragment reads).
// Optional fused ReLU on the A load (input = previous SpMM accumulation).
template <int K, int NOUT, bool RELU>
__global__ __launch_bounds__(128) void gcn_gemm_wmma(const float* __restrict__ A,
                                                     const float* __restrict__ W,
                                                     float* __restrict__ C, int nrows) {
  constexpr int LDA = K + 4;
  __shared__ float As[16 * LDA];
  const int tid = threadIdx.x;
  const int rowBase = blockIdx.x * 16;

  constexpr int F4_PER_ROW = K / 4;
  for (int i = tid; i < 16 * F4_PER_ROW; i += 128) {
    int r = i / F4_PER_ROW;
    int c = (i % F4_PER_ROW) * 4;
    float4 val = *(const float4*)(A + (size_t)(rowBase + r) * K + c);
    if (RELU) {
      val.x = fmaxf(val.x, 0.f); val.y = fmaxf(val.y, 0.f);
      val.z = fmaxf(val.z, 0.f); val.w = fmaxf(val.w, 0.f);
    }
    *(float4*)(&As[r * LDA + c]) = val;
  }
  __syncthreads();

  const int wave = tid >> 5;
  const int lane = tid & 31;
  const int laneM = lane & 15;          // A: row / B: col / C: col
  const int kOff = (lane >> 4) << 1;    // half-wave K offset: 0 or 2
  const int colBase = blockIdx.y * 64 + wave * 16;

  const float* as = &As[laneM * LDA + kOff];
  const float* wp = W + colBase + laneM;

  v8f acc = {};
  for (int k0 = 0; k0 < K; k0 += 4) {
    v2f a = *(const v2f*)(as + k0);                 // A frag: M=laneM, K = k0+kOff, k0+kOff+1
    v2f b;
    b.x = wp[(size_t)(k0 + kOff) * NOUT];           // B frag: N=laneM, K = k0+kOff
    b.y = wp[(size_t)(k0 + kOff + 1) * NOUT];       //                 K = k0+kOff+1
    acc = __builtin_amdgcn_wmma_f32_16x16x4_f32(false, a, false, b, (short)0, acc,
                                                false, false);
  }

  // C layout: VGPR g -> (M = g + 8*(lane>=16), N = lane&15)
  const int row0 = rowBase + ((lane >> 4) << 3);
  const int col = colBase + laneM;
#pragma unroll
  for (int g = 0; g < 8; ++g)
    C[(size_t)(row0 + g) * NOUT + col] = acc[g];
}

// ---------------- Zero fill ----------------
__global__ void gcn_zero_kernel(float* __restrict__ p, size_t n) {
  size_t i = (size_t)blockIdx.x * blockDim.x + threadIdx.x;
  size_t stride = (size_t)gridDim.x * blockDim.x;
  for (; i < n; i += stride) p[i] = 0.f;
}

// ---------------- SpMM: one wave per edge; gather src row, scale, atomic scatter-add ----------------
template <int F>
__global__ void gcn_spmm_kernel(const float* __restrict__ H, const int* __restrict__ src,
                                const int* __restrict__ dst, const float* __restrict__ ew,
                                float* __restrict__ out, int nEdges) {
  int e = blockIdx.x * (blockDim.x >> 5) + (threadIdx.x >> 5);
  if (e >= nEdges) return;
  int lane = threadIdx.x & 31;
  int s = src[e];
  int d = dst[e];
  float w = ew[e];
  const float* hrow = H + (size_t)s * F;
  float* orow = out + (size_t)d * F;
#pragma unroll
  for (int i = 0; i < F / 32; ++i) {
    int f = i * 32 + lane;
    atomicAdd(&orow[f], hrow[f] * w);   // global_atomic_add_f32, L2-resident
  }
}

extern "C" void kernel_launch(void* const* d_in, const int* in_sizes, int n_in,
                              void* d_out, int out_size, void* d_ws, size_t ws_size,
                              hipStream_t stream) {
  const float* x     = (const float*)d_in[0];
  const float* gamma = (const float*)d_in[1];
  const float* beta  = (const float*)d_in[2];
  const float* W1    = (const float*)d_in[3];
  const float* W2    = (const float*)d_in[4];
  const float* W3    = (const float*)d_in[5];
  const float* W4    = (const float*)d_in[6];
  const float* ew    = (const float*)d_in[7];
  const int*   es    = (const int*)d_in[8];
  const int*   ed    = (const int*)d_in[9];
  float* out = (float*)d_out;

  const int n = in_sizes[0] / 256;   // 100000 (multiple of 16: 6250 tiles)
  const int E = in_sizes[7];         // 3.2M edges

  float* ws0 = (float*)d_ws;
  float* ws1 = ws0 + (size_t)n * 256;
  float* ws2 = ws1 + (size_t)n * 256;

  const int lnBlocks = (n + 7) / 8;
  const int spmmBlocks = (E + 7) / 8;

  // h0 = LayerNorm(x)
  gcn_ln_kernel<<<lnBlocks, 256, 0, stream>>>(x, gamma, beta, ws0, n);

  // Layer 1: t = h0 @ W1 (256->256); a = spmm(t)
  gcn_gemm_wmma<256, 256, false><<<dim3(n / 16, 4), 128, 0, stream>>>(ws0, W1, ws1, n);
  gcn_zero_kernel<<<2048, 256, 0, stream>>>(ws2, (size_t)n * 256);
  gcn_spmm_kernel<256><<<spmmBlocks, 256, 0, stream>>>(ws1, es, ed, ew, ws2, E);

  // Layer 2: t = relu(a) @ W2 (256->128); a = spmm(t)
  gcn_gemm_wmma<256, 128, true><<<dim3(n / 16, 2), 128, 0, stream>>>(ws2, W2, ws0, n);
  gcn_zero_kernel<<<2048, 256, 0, stream>>>(ws1, (size_t)n * 128);
  gcn_spmm_kernel<128><<<spmmBlocks, 256, 0, stream>>>(ws0, es, ed, ew, ws1, E);

  // Layer 3: t = relu(a) @ W3 (128->64); a = spmm(t)
  gcn_gemm_wmma<128, 64, true><<<dim3(n / 16, 1), 128, 0, stream>>>(ws1, W3, ws2, n);
  gcn_zero_kernel<<<2048, 256, 0, stream>>>(ws0, (size_t)n * 64);
  gcn_spmm_kernel<64><<<spmmBlocks, 256, 0, stream>>>(ws2, es, ed, ew, ws0, E);

  // Layer 4: t = relu(a) @ W4 (64->64); out = spmm(t)  (no final activation)
  gcn_gemm_wmma<64, 64, true><<<dim3(n / 16, 1), 128, 0, stream>>>(ws0, W4, ws1, n);
  gcn_zero_kernel<<<2048, 256, 0, stream>>>(out, (size_t)n * 64);
  gcn_spmm_kernel<64><<<spmmBlocks, 256, 0, stream>>>(ws1, es, ed, ew, out, E);
}